// ContrasiveLearningLoss_17033840296399
// MI455X (gfx1250) — compile-verified
//
#include <hip/hip_runtime.h>
#include <math.h>

typedef __attribute__((ext_vector_type(2))) float v2f;
typedef __attribute__((ext_vector_type(8))) float v8f;

#define NROWS 8192
#define DDIM  768
#define KCHUNKS 4
#define KCHUNK (DDIM / KCHUNKS)   // 192
#define TILES (NROWS / 16)        // 512

// workspace layout (floats): [0,768) a_colsum | [768,1536) b_colsum | [1536,1536+8192) diag
#define WS_ASUM 0
#define WS_BSUM DDIM
#define WS_DIAG (2 * DDIM)
#define WS_FLOATS (2 * DDIM + NROWS)

// ---------------- init: zero workspace accumulators and output ----------------
__global__ void cl_init_kernel(float* __restrict__ ws, float* __restrict__ out) {
    int i = blockIdx.x * blockDim.x + threadIdx.x;
    if (i < WS_FLOATS) ws[i] = 0.0f;
    if (i == 0) out[0] = 0.0f;
}

// ---------------- column sums of both matrices (coalesced) ----------------
// grid: (DDIM/256, 64) ; each block: 256 columns x 128 rows partial sum
__global__ void cl_colsum_kernel(const float* __restrict__ za,
                                 const float* __restrict__ zb,
                                 float* __restrict__ ws) {
    const int c  = blockIdx.x * 256 + threadIdx.x;      // column 0..767
    const int r0 = blockIdx.y * (NROWS / 64);           // 128-row chunk
    float sa = 0.0f, sb = 0.0f;
    #pragma unroll 4
    for (int r = r0; r < r0 + (NROWS / 64); ++r) {
        sa += za[(size_t)r * DDIM + c];
        sb += zb[(size_t)r * DDIM + c];
    }
    atomicAdd(&ws[WS_ASUM + c], sa);
    atomicAdd(&ws[WS_BSUM + c], sb);
}

// ---------------- per-row dot products via V_WMMA_F32_16X16X4_F32 ----------------
// One wave per (16-row tile, K-chunk).  A-operand: lane L -> row L%16, K-pair 2*(L/16).
// B-operand (= z_b tile transposed) has the mirrored layout: lane L -> col L%16,
// same K-pair split, so both operands load with the identical pattern.
__global__ void cl_diag_wmma_kernel(const float* __restrict__ za,
                                    const float* __restrict__ zb,
                                    float* __restrict__ ws) {
    const int wave   = blockIdx.x * (blockDim.x >> 5) + (threadIdx.x >> 5);
    const int tile   = wave >> 2;          // / KCHUNKS
    const int kchunk = wave & (KCHUNKS - 1);
    const int lane   = threadIdx.x & 31;
    const int row    = tile * 16 + (lane & 15);
    const int koff   = (lane >> 4) * 2;

    const float* pa = za + (size_t)row * DDIM + kchunk * KCHUNK + koff;
    const float* pb = zb + (size_t)row * DDIM + kchunk * KCHUNK + koff;

    v8f c = {0.f, 0.f, 0.f, 0.f, 0.f, 0.f, 0.f, 0.f};
    #pragma unroll 8
    for (int k = 0; k < KCHUNK; k += 4) {
        v2f a = *(const v2f*)(pa + k);   // A[m][kb], A[m][kb+1]
        v2f b = *(const v2f*)(pb + k);   // B[kb][n], B[kb+1][n]  (n = row of z_b tile)
        c = __builtin_amdgcn_wmma_f32_16x16x4_f32(
                /*neg_a=*/false, a, /*neg_b=*/false, b,
                /*c_mod=*/(short)0, c, /*reuse_a=*/false, /*reuse_b=*/false);
    }

    // Diagonal of 16x16 f32 C: VGPR r holds (M=r, N=lane) for lanes 0-15 and
    // (M=r+8, N=lane-16) for lanes 16-31.  Diag lives at lane r (M=r) and
    // lane r+24 (M=r+8).  Extract branchlessly via a cndmask chain, then do
    // ONE predicated atomic instead of 16 branchy regions.
    float dval = 0.0f;
    #pragma unroll
    for (int r = 0; r < 8; ++r) {
        const bool sel = (lane == r) | (lane == r + 24);
        dval = sel ? c[r] : dval;
    }
    const bool isdiag = (lane < 8) | (lane >= 24);
    const int  m      = (lane < 8) ? lane : (lane - 16);   // diag row within tile
    if (isdiag) atomicAdd(&ws[WS_DIAG + tile * 16 + m], dval);
}

// ---------------- sum_i exp(diag_i) - 2*diag_i ----------------
__global__ void cl_diag_finalize_kernel(const float* __restrict__ ws,
                                        float* __restrict__ out) {
    __shared__ float red[256];
    const int i = blockIdx.x * 256 + threadIdx.x;
    const float d = ws[WS_DIAG + i];
    red[threadIdx.x] = expf(d) - 2.0f * d;
    __syncthreads();
    #pragma unroll
    for (int s = 128; s > 0; s >>= 1) {
        if (threadIdx.x < s) red[threadIdx.x] += red[threadIdx.x + s];
        __syncthreads();
    }
    if (threadIdx.x == 0) atomicAdd(out, red[0]);
}

// ---------------- sum(S) = <colsum(z_a), colsum(z_b)> ----------------
__global__ void cl_dot_kernel(const float* __restrict__ ws, float* __restrict__ out) {
    __shared__ float red[256];
    float v = 0.0f;
    for (int i = threadIdx.x; i < DDIM; i += 256)
        v += ws[WS_ASUM + i] * ws[WS_BSUM + i];
    red[threadIdx.x] = v;
    __syncthreads();
    #pragma unroll
    for (int s = 128; s > 0; s >>= 1) {
        if (threadIdx.x < s) red[threadIdx.x] += red[threadIdx.x + s];
        __syncthreads();
    }
    if (threadIdx.x == 0) atomicAdd(out, red[0]);
}

extern "C" void kernel_launch(void* const* d_in, const int* in_sizes, int n_in,
                              void* d_out, int out_size, void* d_ws, size_t ws_size,
                              hipStream_t stream) {
    const float* za = (const float*)d_in[0];
    const float* zb = (const float*)d_in[1];
    float* out = (float*)d_out;
    float* ws  = (float*)d_ws;

    // 1) zero accumulators
    cl_init_kernel<<<(WS_FLOATS + 255) / 256, 256, 0, stream>>>(ws, out);

    // 2) diagonal dots via WMMA (HBM pass; data lands in 192MB L2)
    cl_diag_wmma_kernel<<<(TILES * KCHUNKS) / 8, 256, 0, stream>>>(za, zb, ws);

    // 3) column sums (second pass hits L2)
    dim3 cgrid(DDIM / 256, 64);
    cl_colsum_kernel<<<cgrid, 256, 0, stream>>>(za, zb, ws);

    // 4) reductions
    cl_dot_kernel<<<1, 256, 0, stream>>>(ws, out);
    cl_diag_finalize_kernel<<<NROWS / 256, 256, 0, stream>>>(ws, out);
}